// GraphSAGE_15384572854647
// MI455X (gfx1250) — compile-verified
//
#include <hip/hip_runtime.h>
#include <math.h>

typedef __attribute__((ext_vector_type(2))) float v2f;
typedef __attribute__((ext_vector_type(8))) float v8f;

#define D  128
#define AS 132   // LDS row stride (floats); 132 mod 64 = 4 -> conflict-free fragment reads
#define MT 64    // rows (M) per block: 4 WMMA row-tiles share each B fragment

static __device__ __forceinline__ v8f wmma4(v2f a, v2f b, v8f c) {
    return __builtin_amdgcn_wmma_f32_16x16x4_f32(false, a, false, b, (short)0, c, false, false);
}

// ---------------- utility: on-stream zero fill (graph-capture safe) ----------------
__global__ void gs_zero_i32_kernel(int* __restrict__ p, long n) {
    long i = (long)blockIdx.x * blockDim.x + threadIdx.x;
    long stride = (long)gridDim.x * blockDim.x;
    for (; i < n; i += stride) p[i] = 0;
}

// ---------------- degree histogram: cnt[dst] += 1 (int atomics, cheap) ----------------
__global__ void gs_degree_kernel(const long long* __restrict__ ei, int* __restrict__ cnt, int E) {
    int e = blockIdx.x * blockDim.x + threadIdx.x;
    if (e < E) {
        int dst = (int)ei[(long)E + e];
        atomicAdd(&cnt[dst], 1);
    }
}

// ---------------- single-block exclusive scan: rowptr[0..N] from cnt[0..N) ----------------
__launch_bounds__(1024)
__global__ void gs_scan_kernel(const int* __restrict__ cnt, int* __restrict__ rowptr, int N) {
    __shared__ int part[1024];
    const int tid = threadIdx.x;
    const int C = (N + 1023) / 1024;
    const int start = tid * C;
    const int end = (start + C < N) ? (start + C) : N;
    int s = 0;
    for (int i = start; i < end; ++i) s += cnt[i];
    part[tid] = s;
    __syncthreads();
    for (int off = 1; off < 1024; off <<= 1) {
        int v = (tid >= off) ? part[tid - off] : 0;
        __syncthreads();
        part[tid] += v;
        __syncthreads();
    }
    int base = (tid == 0) ? 0 : part[tid - 1];
    for (int i = start; i < end; ++i) {
        rowptr[i] = base;
        base += cnt[i];
    }
    if (tid == 1023) rowptr[N] = part[1023];
}

// ---------------- bucket fill: CSR neighbor (source) lists ----------------
__global__ void gs_bucket_kernel(const long long* __restrict__ ei,
                                 const int* __restrict__ rowptr,
                                 int* __restrict__ rowcur,
                                 int* __restrict__ esrc, int E) {
    int e = blockIdx.x * blockDim.x + threadIdx.x;
    if (e < E) {
        int src = (int)ei[e];
        int dst = (int)ei[(long)E + e];
        int pos = atomicAdd(&rowcur[dst], 1);
        esrc[rowptr[dst] + pos] = src;
    }
}

// ---------------- per-node insertion sort of neighbor lists -> deterministic sum order ----
__global__ void gs_sortbuckets_kernel(const int* __restrict__ rowptr, int* __restrict__ esrc, int N) {
    int n = blockIdx.x * blockDim.x + threadIdx.x;
    if (n >= N) return;
    const int beg = rowptr[n];
    const int end = rowptr[n + 1];
    for (int i = beg + 1; i < end; ++i) {
        int key = esrc[i];
        int j = i - 1;
        while (j >= beg && esrc[j] > key) { esrc[j + 1] = esrc[j]; --j; }
        esrc[j + 1] = key;
    }
}

// ---------------- CSR gather mean-aggregate: one wave32 per destination node ----------------
__global__ void gs_aggregate_kernel(const float* __restrict__ feat,
                                    const int* __restrict__ rowptr,
                                    const int* __restrict__ esrc,
                                    float* __restrict__ aggm, int N) {
    long gid = (long)blockIdx.x * blockDim.x + threadIdx.x;
    int node = (int)(gid >> 5);
    int lane = (int)(gid & 31);
    if (node >= N) return;
    const int beg = rowptr[node];
    const int end = rowptr[node + 1];
    float4 acc = make_float4(0.f, 0.f, 0.f, 0.f);
    for (int base = beg; base < end; base += 32) {
        int j = base + lane;
        int myid = (j < end) ? esrc[j] : 0;
        int cntb = end - base; if (cntb > 32) cntb = 32;
        for (int jj = 0; jj < cntb; ++jj) {
            int s = __shfl(myid, jj, 32);
            const float4 v = *(const float4*)(feat + (long)s * D + lane * 4);
            acc.x += v.x; acc.y += v.y; acc.z += v.z; acc.w += v.w;
        }
    }
    const float inv = 1.0f / fmaxf((float)(end - beg), 1.0f);
    acc.x *= inv; acc.y *= inv; acc.z *= inv; acc.w *= inv;
    *(float4*)(aggm + (long)node * D + lane * 4) = acc;
}

// ---------------- fused SAGE layer (64-row M-tile): aggm@Wl + b + x@Wr, l2norm, then
//                  mode 0: relu -> out ; mode 1: fused FC(128->2)+softmax -> out2
__launch_bounds__(256)
__global__ void gs_sage_wmma_kernel(const float* __restrict__ aggm,
                                    const float* __restrict__ xin,
                                    const float* __restrict__ Wl,
                                    const float* __restrict__ bl,
                                    const float* __restrict__ Wr,
                                    float* __restrict__ out,
                                    const float* __restrict__ Wfc,
                                    const float* __restrict__ bfc,
                                    float* __restrict__ out2,
                                    int N, int mode) {
    __shared__ float Aagg[MT * AS];   // aggregate tile; reused as output tile
    __shared__ float Ax  [MT * AS];   // self-feature tile
    __shared__ float pp  [MT * 4];
    __shared__ float pq  [MT * 4];
    __shared__ float rn  [MT];

    const int tid = threadIdx.x;
    const int r0  = blockIdx.x * MT;

    // stage A tiles (MT rows x 128), coalesced
    #pragma unroll
    for (int i = 0; i < (MT * 128) / 256; ++i) {       // 32 iterations
        int idx = tid + i * 256;
        int row = idx >> 7;
        int col = idx & 127;
        int gr  = r0 + row;
        float va = 0.0f, vx = 0.0f;
        if (gr < N) {
            va = aggm[(long)gr * D + col];
            vx = xin [(long)gr * D + col];
        }
        Aagg[row * AS + col] = va;
        Ax  [row * AS + col] = vx;
    }
    __syncthreads();

    const int wave = tid >> 5;            // n-tile (column block of 16)
    const int lane = tid & 31;
    const int mn   = lane & 15;
    const int half = lane >> 4;
    const int col  = wave * 16 + mn;      // global output column

    // 4 row-tiles x (Aagg*Wl + X*Wr), B fragments loaded once per k-step
    v8f c0 = {}, c1 = {}, c2 = {}, c3 = {};
    #pragma unroll 2
    for (int kk = 0; kk < 32; ++kk) {
        const int k = kk * 4 + half * 2;
        v2f bL, bR, a;
        bL.x = Wl[(long)k * D + col];  bL.y = Wl[(long)(k + 1) * D + col];
        bR.x = Wr[(long)k * D + col];  bR.y = Wr[(long)(k + 1) * D + col];

        a.x = Aagg[(0 * 16 + mn) * AS + k];  a.y = Aagg[(0 * 16 + mn) * AS + k + 1];
        c0 = wmma4(a, bL, c0);
        a.x = Ax  [(0 * 16 + mn) * AS + k];  a.y = Ax  [(0 * 16 + mn) * AS + k + 1];
        c0 = wmma4(a, bR, c0);

        a.x = Aagg[(1 * 16 + mn) * AS + k];  a.y = Aagg[(1 * 16 + mn) * AS + k + 1];
        c1 = wmma4(a, bL, c1);
        a.x = Ax  [(1 * 16 + mn) * AS + k];  a.y = Ax  [(1 * 16 + mn) * AS + k + 1];
        c1 = wmma4(a, bR, c1);

        a.x = Aagg[(2 * 16 + mn) * AS + k];  a.y = Aagg[(2 * 16 + mn) * AS + k + 1];
        c2 = wmma4(a, bL, c2);
        a.x = Ax  [(2 * 16 + mn) * AS + k];  a.y = Ax  [(2 * 16 + mn) * AS + k + 1];
        c2 = wmma4(a, bR, c2);

        a.x = Aagg[(3 * 16 + mn) * AS + k];  a.y = Aagg[(3 * 16 + mn) * AS + k + 1];
        c3 = wmma4(a, bL, c3);
        a.x = Ax  [(3 * 16 + mn) * AS + k];  a.y = Ax  [(3 * 16 + mn) * AS + k + 1];
        c3 = wmma4(a, bR, c3);
    }
    __syncthreads();                      // done reading Aagg; reuse it as the output tile

    // spill C fragments (+bias): VGPR v <-> row v (lanes 0-15) / v+8 (lanes 16-31)
    const float bias = bl[col];
    #pragma unroll
    for (int v = 0; v < 8; ++v) {
        const int rr = v + half * 8;
        Aagg[(0 * 16 + rr) * AS + col] = c0[v] + bias;
        Aagg[(1 * 16 + rr) * AS + col] = c1[v] + bias;
        Aagg[(2 * 16 + rr) * AS + col] = c2[v] + bias;
        Aagg[(3 * 16 + rr) * AS + col] = c3[v] + bias;
    }
    __syncthreads();

    // per-row sum of squares (4 segments of 32 per row)
    {
        int row = tid >> 2;
        int seg = tid & 3;
        float s = 0.0f;
        #pragma unroll
        for (int j = 0; j < 32; ++j) {
            float t = Aagg[row * AS + seg * 32 + j];
            s += t * t;
        }
        pp[row * 4 + seg] = s;
    }
    __syncthreads();
    if (tid < MT) {
        float s = pp[tid * 4 + 0] + pp[tid * 4 + 1] + pp[tid * 4 + 2] + pp[tid * 4 + 3];
        rn[tid] = 1.0f / fmaxf(sqrtf(s), 1e-12f);
    }
    __syncthreads();

    if (mode == 0) {
        // l2-normalize + relu, write hidden features
        #pragma unroll
        for (int i = 0; i < (MT * 128) / 256; ++i) {
            int idx  = tid + i * 256;
            int row  = idx >> 7;
            int ccol = idx & 127;
            int gr   = r0 + row;
            if (gr < N) {
                float vv = Aagg[row * AS + ccol] * rn[row];
                out[(long)gr * D + ccol] = fmaxf(vv, 0.0f);
            }
        }
    } else {
        // l2-normalize, then fused FC(128->2) + softmax
        {
            int row = tid >> 2;
            int seg = tid & 3;
            float inv = rn[row];
            float p0 = 0.0f, p1 = 0.0f;
            #pragma unroll
            for (int j = 0; j < 32; ++j) {
                int k   = seg * 32 + j;
                float h = Aagg[row * AS + k] * inv;
                p0 += h * Wfc[k * 2 + 0];
                p1 += h * Wfc[k * 2 + 1];
            }
            pp[row * 4 + seg] = p0;
            pq[row * 4 + seg] = p1;
        }
        __syncthreads();
        if (tid < MT) {
            int gr = r0 + tid;
            if (gr < N) {
                float l0 = bfc[0] + pp[tid * 4 + 0] + pp[tid * 4 + 1] + pp[tid * 4 + 2] + pp[tid * 4 + 3];
                float l1 = bfc[1] + pq[tid * 4 + 0] + pq[tid * 4 + 1] + pq[tid * 4 + 2] + pq[tid * 4 + 3];
                float m  = fmaxf(l0, l1);
                float e0 = expf(l0 - m);
                float e1 = expf(l1 - m);
                float s  = e0 + e1;
                out2[(long)gr * 2 + 0] = e0 / s;
                out2[(long)gr * 2 + 1] = e1 / s;
            }
        }
    }
}

extern "C" void kernel_launch(void* const* d_in, const int* in_sizes, int n_in,
                              void* d_out, int out_size, void* d_ws, size_t ws_size,
                              hipStream_t stream) {
    const float*     x   = (const float*)d_in[0];
    const long long* ei  = (const long long*)d_in[1];   // int64 edge_index, shape (2,E)
    const float*     W1l = (const float*)d_in[2];
    const float*     b1l = (const float*)d_in[3];
    const float*     W1r = (const float*)d_in[4];
    const float*     W2l = (const float*)d_in[5];
    const float*     b2l = (const float*)d_in[6];
    const float*     W2r = (const float*)d_in[7];
    const float*     Wfc = (const float*)d_in[8];
    const float*     bfc = (const float*)d_in[9];
    float* out = (float*)d_out;

    const int N = in_sizes[0] / D;
    const int E = in_sizes[1] / 2;

    // workspace layout (all 4-byte elements):
    // [cnt: N int][rowcur: N int][rowptr: N+1 int][esrc: E int][agg: N*D f32][h1: N*D f32]
    int*   cnt    = (int*)d_ws;
    int*   rowcur = cnt + N;
    int*   rowptr = rowcur + N;
    int*   esrc   = rowptr + (N + 1);
    float* aggb   = (float*)(esrc + E);
    float* h1     = aggb + (size_t)N * D;

    const int eb  = (E + 255) / 256;                      // one thread per edge
    const int nb  = (N + 255) / 256;                      // one thread per node
    const int agg_blocks  = (N + 7) / 8;                  // one wave32 per node, 8/block
    const int sage_blocks = (N + MT - 1) / MT;

    // -------- build CSR once (shared by both layers), deterministic order --------
    gs_zero_i32_kernel<<<512, 256, 0, stream>>>(cnt, 2L * N);     // cnt + rowcur
    gs_degree_kernel<<<eb, 256, 0, stream>>>(ei, cnt, E);
    gs_scan_kernel<<<1, 1024, 0, stream>>>(cnt, rowptr, N);
    gs_bucket_kernel<<<eb, 256, 0, stream>>>(ei, rowptr, rowcur, esrc, E);
    gs_sortbuckets_kernel<<<nb, 256, 0, stream>>>(rowptr, esrc, N);

    // -------- layer 1 --------
    gs_aggregate_kernel<<<agg_blocks, 256, 0, stream>>>(x, rowptr, esrc, aggb, N);
    gs_sage_wmma_kernel<<<sage_blocks, 256, 0, stream>>>(
        aggb, x, W1l, b1l, W1r, h1, nullptr, nullptr, nullptr, N, 0);

    // -------- layer 2 (FC + softmax fused) --------
    gs_aggregate_kernel<<<agg_blocks, 256, 0, stream>>>(h1, rowptr, esrc, aggb, N);
    gs_sage_wmma_kernel<<<sage_blocks, 256, 0, stream>>>(
        aggb, h1, W2l, b2l, W2r, nullptr, Wfc, bfc, out, N, 1);
}